// DigitCaps_5695126634746
// MI455X (gfx1250) — compile-verified
//
#include <hip/hip_runtime.h>
#include <math.h>

typedef __attribute__((ext_vector_type(2))) float v2f;
typedef __attribute__((ext_vector_type(8))) float v8f;

#define R_TOT   1152
#define I_DIM   8
#define C_DIM   10
#define O_DIM   16
#define CO      160          // C*O
#define WAVES   8
#define TPB     256          // 8 wave32
#define R_PER_WAVE (R_TOT / WAVES)   // 144

// ds_swizzle_b32 immediate: group-of-32, and_mask=0x1F, or=0, xor=0x10 (SWAPX16)
#define SWZ_SWAPX16 0x401F

// D = A(16x4 f32) * B(4x16 f32) + C(16x16 f32)
static __device__ __forceinline__ v8f wmma4(v2f a, v2f b, v8f c) {
  return __builtin_amdgcn_wmma_f32_16x16x4_f32(false, a, false, b,
                                               (short)0, c, false, false);
}

// One r-sweep of the routing pass, branch-free (unswitched on FIRST).
// Transposed WMMA tile: u'[o, m] = sum_i W[r,i,c,o] * x[b0+m, r, i]
//   A-frag (16x4): lane l -> row o = l&15, k = 2*(l>>4)+j  (W elements)
//   B-frag (4x16): lane l -> col m = l&15, k = 2*(l>>4)+j  (x elements)
//   D (16x16):     lane l -> col m = l&15, rows o = j + 8*(l>>4)
// Routing logit t[m,c] = <u_hat[m,c,:], Vsum[m,c,:]> is 8 lane-local FMAs +
// one ds_swizzle (SWAPX16) to combine the o=0..7 / 8..15 halves, and
// coef = exp(t) lands lane-local exactly where the s-accumulate needs it.
template <bool FIRST>
static __device__ __forceinline__ void sweep(
    const float* __restrict__ xrow, const float* __restrict__ wbase,
    const float* __restrict__ Vsum, int rBeg, int rEnd, int mm, int hh,
    v8f sreg[C_DIM], float dreg[C_DIM])
{
  const float* vbase = Vsum + (mm * C_DIM) * O_DIM + hh * 8;
  for (int r = rBeg; r < rEnd; ++r) {
    const float* wr = wbase + (size_t)r * (I_DIM * CO);
    const float* xr = xrow  + (size_t)r * I_DIM;
    __builtin_prefetch(wr + I_DIM * CO, 0, 1);   // next-r W row -> global_prefetch

    // x B-fragments, shared by all 10 capsules: i = kk*4 + 2*hh + {0,1}
    v2f bx0 = *(const v2f*)(xr);
    v2f bx1 = *(const v2f*)(xr + 4);

    #pragma unroll
    for (int c = 0; c < C_DIM; ++c) {
      // W A-fragments (transposed W): row o=mm, i = kk*4 + 2*hh + j
      v2f aw0, aw1;
      aw0.x = wr[0 * CO + c * O_DIM];
      aw0.y = wr[1 * CO + c * O_DIM];
      aw1.x = wr[4 * CO + c * O_DIM];
      aw1.y = wr[5 * CO + c * O_DIM];

      v8f u = {};
      u = wmma4(aw0, bx0, u);   // K = 0..3
      u = wmma4(aw1, bx1, u);   // K = 4..7 -> u'[o, m] tile for capsule c

      if (FIRST) {
        // softmax of zeros is uniform: coef = 1, normalize by R at pass end
        #pragma unroll
        for (int j = 0; j < 8; ++j) sreg[c][j] += u[j];
      } else {
        const float* vsp = vbase + c * O_DIM;
        float pd = 0.0f;
        #pragma unroll
        for (int j = 0; j < 8; ++j) pd += u[j] * vsp[j];
        float t = pd + __int_as_float(
            __builtin_amdgcn_ds_swizzle(__float_as_int(pd), SWZ_SWAPX16));
        float coef = __expf(t);
        dreg[c] += coef;
        #pragma unroll
        for (int j = 0; j < 8; ++j) sreg[c][j] += coef * u[j];
      }
    }
  }
}

__global__ __launch_bounds__(TPB) void digitcaps_routing(
    const float* __restrict__ x,     // (B, R, I)
    const float* __restrict__ W,     // (R, I, C, O)
    float* __restrict__ out)         // (B, C, O)
{
  __shared__ float sAcc[16 * C_DIM * O_DIM];  // [(m*10+c)*16 + o] unnormalized s
  __shared__ float dAcc[16 * C_DIM];          // [m*10+c]          softmax denom
  __shared__ float Vsum[16 * C_DIM * O_DIM];  // [(m*10+c)*16 + o] v0+v1 running sum

  const int tid  = threadIdx.x;
  const int wave = tid >> 5;
  const int lane = tid & 31;
  const int hh   = lane >> 4;   // half of wave
  const int mm   = lane & 15;
  const int b0   = blockIdx.x << 4;

  for (int p = tid; p < 16 * CO; p += TPB) Vsum[p] = 0.0f;

  const int rBeg = wave * R_PER_WAVE;
  const int rEnd = rBeg + R_PER_WAVE;

  // per-lane base pointers (fold lane-dependent parts once)
  const float* xrow  = x + (size_t)(b0 + mm) * (R_TOT * I_DIM) + hh * 2;
  const float* wbase = W + (size_t)hh * (2 * CO) + mm;

  for (int pass = 0; pass < 3; ++pass) {
    for (int p = tid; p < 16 * CO; p += TPB)    sAcc[p] = 0.0f;
    for (int p = tid; p < 16 * C_DIM; p += TPB) dAcc[p] = 0.0f;
    __syncthreads();

    v8f   sreg[C_DIM];
    float dreg[C_DIM];
    #pragma unroll
    for (int c = 0; c < C_DIM; ++c) {
      dreg[c] = 0.0f;
      #pragma unroll
      for (int j = 0; j < 8; ++j) sreg[c][j] = 0.0f;
    }

    if (pass == 0)
      sweep<true >(xrow, wbase, Vsum, rBeg, rEnd, mm, hh, sreg, dreg);
    else
      sweep<false>(xrow, wbase, Vsum, rBeg, rEnd, mm, hh, sreg, dreg);

    // merge wave partials into LDS
    #pragma unroll
    for (int c = 0; c < C_DIM; ++c) {
      #pragma unroll
      for (int j = 0; j < 8; ++j)
        atomicAdd(&sAcc[(mm * C_DIM + c) * O_DIM + hh * 8 + j], sreg[c][j]);
    }
    if (pass != 0 && hh == 0) {   // both halves hold identical dreg; write one
      #pragma unroll
      for (int c = 0; c < C_DIM; ++c)
        atomicAdd(&dAcc[mm * C_DIM + c], dreg[c]);
    }
    __syncthreads();

    // normalize + squash (one thread per (m,c))
    if (tid < 16 * C_DIM) {
      const int m = tid / C_DIM, c = tid % C_DIM;
      const float invd = (pass == 0) ? (1.0f / (float)R_TOT)
                                     : 1.0f / dAcc[m * C_DIM + c];
      float sv[O_DIM];
      float nrm = 0.0f;
      #pragma unroll
      for (int o = 0; o < O_DIM; ++o) {
        float v = sAcc[(m * C_DIM + c) * O_DIM + o] * invd;
        sv[o] = v;
        nrm += v * v;
      }
      const float scale = nrm / ((nrm + 1.0f) * sqrtf(fmaxf(nrm, 1e-30f)));
      if (pass < 2) {
        #pragma unroll
        for (int o = 0; o < O_DIM; ++o)
          Vsum[(m * C_DIM + c) * O_DIM + o] += scale * sv[o];
      } else {
        #pragma unroll
        for (int o = 0; o < O_DIM; ++o)
          out[(size_t)(b0 + m) * CO + c * O_DIM + o] = scale * sv[o];
      }
    }
    __syncthreads();
  }
}

extern "C" void kernel_launch(void* const* d_in, const int* in_sizes, int n_in,
                              void* d_out, int out_size, void* d_ws, size_t ws_size,
                              hipStream_t stream) {
  const float* x   = (const float*)d_in[0];   // (1024, 1152, 8)   fp32
  const float* W   = (const float*)d_in[1];   // (1152, 8, 10, 16) fp32
  float*       out = (float*)d_out;           // (1024, 10, 16)    fp32
  (void)in_sizes; (void)n_in; (void)out_size; (void)d_ws; (void)ws_size;
  digitcaps_routing<<<64, TPB, 0, stream>>>(x, W, out);
}